// YOLOv3LossOriginal_17145509445936
// MI455X (gfx1250) — compile-verified
//
#include <hip/hip_runtime.h>
#include <math.h>

// ---------------------------------------------------------------------------
// YOLOv3 loss for MI455X (gfx1250).
// Key insight: IGNORE_THRESH == TRUTH_THRESH, so only the conf channel is
// needed for ~all cells; full 85-channel work only at <=1600 scattered obj
// cells. Total required traffic ~1.5 MB (not 88 MB) -> latency-bound tiny
// kernel. Conf planes are staged via the Tensor Data Mover (async, TENSORcnt)
// and the final cross-lane reduction uses V_WMMA_F32_16X16X4_F32.
// ---------------------------------------------------------------------------

typedef unsigned int uint32x4 __attribute__((ext_vector_type(4)));
typedef int          int32x4  __attribute__((ext_vector_type(4)));
typedef int          int32x8  __attribute__((ext_vector_type(8)));
typedef float        v2f      __attribute__((ext_vector_type(2)));
typedef float        v8f      __attribute__((ext_vector_type(8)));

#define IGNORE_THRESH 0.5f
#define OBJECT_SCALE  5.0f
#define NUM_CLASSES   80

struct Rec { float tx, ty, tw, th; int cls; int pad; };

__device__ __forceinline__ float sigm(float z) { return 1.0f / (1.0f + expf(-z)); }
__device__ __forceinline__ float clog(float p) { return fmaxf(logf(p), -100.0f); }

// --- kernel 1: zero output, poison map with -1 ------------------------------
__global__ void yolo_init_kernel(int* __restrict__ map, float* __restrict__ out, int ncells) {
    int i = blockIdx.x * blockDim.x + threadIdx.x;
    if (i < ncells) map[i] = -1;
    if (i == 0) out[0] = 0.0f;
}

// --- kernel 2: build target records + scatter map (1 thread per batch) ------
__global__ void yolo_build_kernel(const float* __restrict__ target,
                                  const float* __restrict__ anchors,
                                  int* __restrict__ map, Rec* __restrict__ recs,
                                  int B, int T, int A, int G) {
    int b = blockIdx.x * blockDim.x + threadIdx.x;
    if (b >= B) return;
    for (int t = 0; t < T; ++t) {                 // ascending t: last write wins
        const float* tg = target + (size_t)(b * T + t) * 5;
        float cls = tg[0], x = tg[1], y = tg[2], w = tg[3], h = tg[4];
        if ((cls + x + y + w + h) == 0.0f) continue;          // invalid row
        float gx = x * (float)G, gy = y * (float)G;
        float gw = w * (float)G, gh = h * (float)G;
        int gi = (int)gx, gj = (int)gy;
        float area_t = (gw + 1.0f) * (gh + 1.0f);
        int best = 0; float best_iou = -1.0f;
        for (int a = 0; a < A; ++a) {             // first max wins (argmax)
            float aw = anchors[a * 2 + 0], ah = anchors[a * 2 + 1];
            float inter  = (fminf(gw, aw) + 1.0f) * (fminf(gh, ah) + 1.0f);
            float area_a = (aw + 1.0f) * (ah + 1.0f);
            float iou = inter / (area_t + area_a - inter + 1e-16f);
            if (iou > best_iou) { best_iou = iou; best = a; }
        }
        if (!(best_iou > IGNORE_THRESH)) continue;            // ig == tr set
        if (gi < 0 || gi >= G || gj < 0 || gj >= G) continue; // mode='drop'
        int r = b * T + t;
        Rec rc;
        rc.tx = gx - (float)gi;
        rc.ty = gy - (float)gj;
        rc.tw = logf(gw / anchors[best * 2 + 0] + 1e-16f);
        rc.th = logf(gh / anchors[best * 2 + 1] + 1e-16f);
        rc.cls = (int)cls; rc.pad = 0;
        recs[r] = rc;
        map[((b * A + best) * G + gj) * G + gi] = r;
    }
}

// --- kernel 3: per-plane loss; TDM stage conf plane -> LDS; WMMA reduce -----
__global__ void yolo_loss_kernel(const float* __restrict__ pred,
                                 const int* __restrict__ map,
                                 const Rec* __restrict__ recs,
                                 float* __restrict__ out, int G) {
    __shared__ float splane[52 * 52];             // one conf plane (10.8 KB of 320 KB)
    const int plane = blockIdx.x;                 // plane = b*A + a
    const int P = G * G;
    const float* ch = pred + (size_t)plane * 85 * P;   // channel base for (b,a)
    const float* conf_plane = ch + (size_t)4 * P;      // channel 4 = conf logit

    // Wave 0 issues one TDM load of the whole conf plane into LDS.
    if (threadIdx.x < 32) {
        unsigned lds_addr = (unsigned)(size_t)&splane[0];   // flat low 32b == LDS byte offset
        unsigned long long ga = (unsigned long long)(size_t)conf_plane;
        uint32x4 g0;
        g0.x = 1u;                                          // count=1 (valid descriptor)
        g0.y = lds_addr;                                    // lds_addr[31:0]
        g0.z = (unsigned)(ga & 0xFFFFFFFFull);              // global_addr[31:0]
        g0.w = (unsigned)((ga >> 32) & 0x1FFFFFFull)        // global_addr[56:32]
             | (2u << 30);                                  // type=2 ("image")
        int32x8 g1;
        g1[0] = (int)(2u << 16);                            // data_size=2 -> 4 bytes
        g1[1] = (int)(((unsigned)P & 0xFFFFu) << 16);       // tensor_dim0[15:0]
        g1[2] = (int)(((unsigned)P >> 16) | (1u << 16));    // tensor_dim0[31:16], tensor_dim1=1
        g1[3] = (int)(((unsigned)P & 0xFFFFu) << 16);       // tile_dim0 = P (multiple of 4B)
        g1[4] = 1;                                          // tile_dim1=1, tile_dim2=0
        g1[5] = P;                                          // tensor_dim0_stride[31:0]
        g1[6] = 0;                                          // stride0[47:32], stride1[15:0]
        g1[7] = 0;                                          // stride1[47:16]
        int32x4 gz = {0, 0, 0, 0};                          // groups 2/3 unused (<=2D)
#if defined(__clang_major__) && __clang_major__ >= 23
        int32x8 gz8 = {0, 0, 0, 0, 0, 0, 0, 0};
        __builtin_amdgcn_tensor_load_to_lds(g0, g1, gz, gz, gz8, 0);
#else
        __builtin_amdgcn_tensor_load_to_lds(g0, g1, gz, gz, 0);
#endif
        __builtin_amdgcn_s_wait_tensorcnt(0);               // s_wait_tensorcnt 0
    }
    __syncthreads();

    float acc = 0.0f;
    for (int idx = threadIdx.x; idx < P; idx += blockDim.x) {
        float conf = sigm(splane[idx]);
        int r = map[plane * P + idx];
        if (r < 0) {
            // noobj: NOOBJECT_SCALE * bce(conf, 0)
            acc += -clog(1.0f - conf);
        } else {
            Rec rc = recs[r];
            // coord MSE (COORD_SCALE = 1); w,h use raw logits per reference
            float px = sigm(ch[(size_t)0 * P + idx]);
            float py = sigm(ch[(size_t)1 * P + idx]);
            float pw = ch[(size_t)2 * P + idx];
            float ph = ch[(size_t)3 * P + idx];
            float dx = px - rc.tx, dy = py - rc.ty, dw = pw - rc.tw, dh = ph - rc.th;
            acc += dx * dx + dy * dy + dw * dw + dh * dh;
            // OBJECT_SCALE * bce(conf, 1)
            acc += OBJECT_SCALE * (-clog(conf));
            // class BCE vs one-hot
            for (int c = 0; c < NUM_CLASSES; ++c) {
                float pc = sigm(ch[(size_t)(5 + c) * P + idx]);
                float lp = clog(pc), l1 = clog(1.0f - pc);
                float tt = (c == rc.cls) ? 1.0f : 0.0f;
                acc += -(tt * lp + (1.0f - tt) * l1);
            }
        }
    }

    // Wave-level reduction via one V_WMMA_F32_16X16X4_F32:
    // A[m][k] = {acc_m, 0} (lanes 0-15 -> K0/K1, lanes 16-31 -> K2/K3), B = ones
    // => D[m][n] = acc_m + acc_{m+16}. Lane 0 VGPRs hold rowsums 0..7,
    // lane 16 VGPRs hold rowsums 8..15. EXEC is all-1s here (reconverged).
    v2f a; a.x = acc;  a.y = 0.0f;
    v2f bm; bm.x = 1.0f; bm.y = 1.0f;
    v8f cz = {};
    v8f d = __builtin_amdgcn_wmma_f32_16x16x4_f32(
        /*neg_a=*/false, a, /*neg_b=*/false, bm,
        /*c_mod=*/(short)0, cz, /*reuse_a=*/false, /*reuse_b=*/false);
    float s = d[0] + d[1] + d[2] + d[3] + d[4] + d[5] + d[6] + d[7];
    int lane = threadIdx.x & 31;
    if (lane == 0 || lane == 16) atomicAdd(out, s);
}

extern "C" void kernel_launch(void* const* d_in, const int* in_sizes, int n_in,
                              void* d_out, int out_size, void* d_ws, size_t ws_size,
                              hipStream_t stream) {
    (void)n_in; (void)out_size; (void)ws_size;
    const float* pred    = (const float*)d_in[0];
    const float* target  = (const float*)d_in[1];
    const float* anchors = (const float*)d_in[2];
    // d_in[3] = num_anchors, d_in[4] = grid_size (device scalars; derived from sizes instead)

    const int A = in_sizes[2] / 2;                       // anchors: (A,2)
    const int T = 50;                                    // targets per batch (reference)
    const int B = in_sizes[1] / (T * 5);                 // target: (B,T,5)
    const long long g2 = (long long)in_sizes[0] / ((long long)B * A * (5 + NUM_CLASSES));
    const int G = (int)(sqrt((double)g2) + 0.5);         // pred: (B, A*85, G, G)
    const int planes = B * A;
    const int ncells = planes * G * G;

    int*  map  = (int*)d_ws;
    size_t rec_off = (((size_t)ncells * sizeof(int)) + 255) & ~(size_t)255;
    Rec*  recs = (Rec*)((char*)d_ws + rec_off);
    float* out = (float*)d_out;

    yolo_init_kernel<<<(ncells + 255) / 256, 256, 0, stream>>>(map, out, ncells);
    yolo_build_kernel<<<1, 64, 0, stream>>>(target, anchors, map, recs, B, T, A, G);
    yolo_loss_kernel<<<planes, 256, 0, stream>>>(pred, map, recs, out, G);
}